// AdEx_5428838662825
// MI455X (gfx1250) — compile-verified
//
#include <hip/hip_runtime.h>

// AdEx Euler scan: N=1024 neurons, T=40000 serially-dependent steps.
// Latency-bound recurrence. Critical chain: fma -> v_exp_f32 -> fma -> cndmask
// (~4 dep ops/step, ~12 issued VALU/step); 32 wave32 waves spread 1-per-SIMD
// (8 blocks x 128). Output streaming (327 MB) staged per-wave in LDS and
// drained with gfx1250 async LDS->global b128 stores (ASYNCcnt), double-buffered.

#define T_STEPS 40000
#define NN      1024
#define TILE    16            // timesteps per LDS tile
#define BLK     128           // threads per block
#define WAVES   (BLK / 32)    // waves per block

typedef int v4i __attribute__((ext_vector_type(4)));
typedef __attribute__((address_space(1))) v4i g_v4i;
typedef __attribute__((address_space(3))) v4i l_v4i;

__device__ __forceinline__ void async_store_b128(float* gptr, float* lptr) {
#if __has_builtin(__builtin_amdgcn_global_store_async_from_lds_b128)
    __builtin_amdgcn_global_store_async_from_lds_b128(
        (g_v4i*)gptr, (l_v4i*)lptr, /*offset=*/0, /*cpol=*/0);
#else
    unsigned long long ga = (unsigned long long)gptr;
    unsigned int       la = (unsigned int)(unsigned long long)lptr; // low 32 bits = LDS address
    asm volatile("global_store_async_from_lds_b128 %0, %1, off"
                 :: "v"(ga), "v"(la) : "memory");
#endif
}

__device__ __forceinline__ float fast_exp2(float x) {
#if __has_builtin(__builtin_amdgcn_exp2f)
    return __builtin_amdgcn_exp2f(x);   // raw v_exp_f32
#else
    return exp2f(x);
#endif
}

__global__ __launch_bounds__(BLK)
void adex_scan_kernel(const float* __restrict__ I_ext,
                      const float* __restrict__ V0,
                      const float* __restrict__ w0,
                      const float* __restrict__ pV_rest,
                      const float* __restrict__ pV_reset,
                      const float* __restrict__ pV_T,
                      const float* __restrict__ pV_thres,
                      const float* __restrict__ pdelta_T,
                      const float* __restrict__ pR,
                      const float* __restrict__ ptau,
                      const float* __restrict__ ptau_w,
                      const float* __restrict__ pa,
                      const float* __restrict__ pb,
                      float* __restrict__ outV,
                      float* __restrict__ outW)
{
    // Wave-private double-buffered staging tiles: [buf][wave][step][lane].
    __shared__ float vbuf[2][WAVES][TILE][32];
    __shared__ float wbuf[2][WAVES][TILE][32];

    const int tid  = threadIdx.x;
    const int wave = tid >> 5;
    const int lane = tid & 31;
    const int i    = blockIdx.x * BLK + tid;          // neuron id

    // Uniform scalar parameters -> SGPRs; fold constants to shorten the chain.
    const float dt        = 5e-5f;
    const float V_rest    = *pV_rest;
    const float V_reset   = *pV_reset;                // spike target (== V+dt*((V_reset-V)/dt) to ~1 ulp)
    const float V_T       = *pV_T;
    const float V_thres   = *pV_thres;
    const float delta_T   = *pdelta_T;
    const float R         = *pR;
    const float inv_tau   = 1.0f / *ptau;
    const float inv_tau_w = 1.0f / *ptau_w;
    const float a         = *pa;
    const float b         = *pb;

    const float LOG2E = 1.44269504088896340736f;
    const float e1   = (1.0f / delta_T) * LOG2E;      // exp((V-V_T)/dT) = 2^(V*e1 + e0)
    const float e0   = -V_T * (1.0f / delta_T) * LOG2E;
    const float c    = dt * inv_tau;                  // dt/tau
    const float cdT  = c * delta_T;
    const float cR   = c * R;
    const float cw   = dt * inv_tau_w;                // dt/tau_w
    const float ncwa = -(cw * a);                     // applied to (V_rest - V)
    const float omcw = 1.0f - cw;

    float V = V0[i];
    float w = w0[i];

    const float4* __restrict__ I4 = reinterpret_cast<const float4*>(I_ext);

    // Flush lane mapping: each async b128 covers 4 tile-rows x 128B.
    const int r  = lane >> 3;          // row within 4-row chunk
    const int c4 = (lane & 7) << 2;    // float column within 32-neuron row
    const int nb = blockIdx.x * BLK + wave * 32;      // wave's neuron base

    for (int kt = 0; kt < T_STEPS / TILE; ++kt) {
        const int p     = kt & 1;
        const int kbase = kt * TILE;

        // Buffer p was flushed 2 tiles ago; async stores complete in order,
        // so <=8 outstanding means only the previous tile's flush remains.
        asm volatile("s_wait_asynccnt 0x8" ::: "memory");

        #pragma unroll
        for (int q4 = 0; q4 < TILE / 4; ++q4) {
            const float4 Iq = I4[(kbase >> 2) + q4];
            #pragma unroll
            for (int u4 = 0; u4 < 4; ++u4) {
                const int   u  = q4 * 4 + u4;
                const float Ik = (u4 == 0) ? Iq.x : (u4 == 1) ? Iq.y
                               : (u4 == 2) ? Iq.z : Iq.w;

                // Emit pre-step state into the staging tile.
                vbuf[p][wave][u][lane] = V;
                wbuf[p][wave][u][lane] = w;

                // Critical chain: fma -> v_exp_f32 -> fma -> cndmask.
                const float E   = fast_exp2(fmaf(V, e1, e0));
                // Off-chain (computed in parallel with the exp):
                const float Vrd = V_rest - V;
                float base = fmaf(c,  Vrd,    V);
                base       = fmaf(cR, Ik - w, base);
                const float Vn = fmaf(cdT, E, base);
                const bool  spike = V > V_thres;

                // Adaptation update (parallel to V chain); reuses Vrd.
                const float wn = fmaf(ncwa, Vrd, omcw * w);

                V = spike ? V_reset  : Vn;
                w = spike ? (wn + b) : wn;
            }
        }

        // Make ds_stores visible before the async engine reads LDS.
        asm volatile("s_wait_dscnt 0x0" ::: "memory");

        // Flush tile: 4 async b128 ops per array (512B each, fully coalesced).
        #pragma unroll
        for (int q = 0; q < 4; ++q) {
            const int u = q * 4 + r;
            const int k = kbase + u;
            async_store_b128(&outV[(size_t)k * NN + nb + c4], &vbuf[p][wave][u][c4]);
            async_store_b128(&outW[(size_t)k * NN + nb + c4], &wbuf[p][wave][u][c4]);
        }
    }

    asm volatile("s_wait_asynccnt 0x0" ::: "memory");
}

extern "C" void kernel_launch(void* const* d_in, const int* in_sizes, int n_in,
                              void* d_out, int out_size, void* d_ws, size_t ws_size,
                              hipStream_t stream) {
    (void)in_sizes; (void)n_in; (void)out_size; (void)d_ws; (void)ws_size;

    const float* I_ext  = (const float*)d_in[0];
    const float* V0     = (const float*)d_in[1];
    const float* w0     = (const float*)d_in[2];
    const float* V_rest = (const float*)d_in[3];
    const float* V_rst  = (const float*)d_in[4];
    const float* V_T    = (const float*)d_in[5];
    const float* V_thr  = (const float*)d_in[6];
    const float* dT     = (const float*)d_in[7];
    const float* R      = (const float*)d_in[8];
    const float* tau    = (const float*)d_in[9];
    const float* tau_w  = (const float*)d_in[10];
    const float* a      = (const float*)d_in[11];
    const float* b      = (const float*)d_in[12];

    float* outV = (float*)d_out;                        // Vs: [T_STEPS, NN]
    float* outW = (float*)d_out + (size_t)T_STEPS * NN; // ws: [T_STEPS, NN]

    // 32 waves spread 1-per-SIMD across 32 SIMDs (8 WGPs).
    adex_scan_kernel<<<dim3(NN / BLK), dim3(BLK), 0, stream>>>(
        I_ext, V0, w0, V_rest, V_rst, V_T, V_thr, dT, R, tau, tau_w, a, b,
        outV, outW);
}